// Oracle_30159260352676
// MI455X (gfx1250) — compile-verified
//
#include <hip/hip_runtime.h>
#include <math.h>

// MI455X / gfx1250, wave32. fp32 matrix work -> V_WMMA_F32_16X16X4_F32.
//
// Algebra: lam <- relu(lam + eta*(c - lam@G)),  c = o@R - r,  G = R^T R.
// G (256x256 f32 = 256KB) lives in LDS; iteration loop touches no global mem.
// lam is kept BOTH in registers (C-layout, for the pointwise update) and in
// per-wave LDS (row-major, feeding A-fragments) -- LDS is only the layout
// exchange. Final y = o - lam@R^T computed once against pre-transposed R.

typedef float v2f __attribute__((ext_vector_type(2)));
typedef float v4f __attribute__((ext_vector_type(4)));
typedef float v8f __attribute__((ext_vector_type(8)));
typedef int   v4i_g __attribute__((vector_size(4 * sizeof(int))));  // matches builtin param

#define B_ROWS 16384
#define D_IN   1024
#define M_IN   512
#define D_OUT  512
#define M_OUT  256
#define PROJ_ITERS 50

#if defined(__has_builtin)
#  if __has_builtin(__builtin_amdgcn_global_load_async_to_lds_b128)
#    define HAVE_ASYNC_LDS 1
#  endif
#endif
#ifndef HAVE_ASYNC_LDS
#  define HAVE_ASYNC_LDS 0
#endif

// D = A(16x4) * B(4x16) + C, fp32. Immediates written literally at the builtin.
__device__ __forceinline__ v8f wmma4(v2f a, v2f b, v8f c) {
  return __builtin_amdgcn_wmma_f32_16x16x4_f32(false, a, false, b, (short)0, c,
                                               false, false);
}

// ---------------------------------------------------------------- prep kernels
__global__ void k_transpose_R(const float* __restrict__ R, float* __restrict__ Rt) {
  int idx = blockIdx.x * blockDim.x + threadIdx.x;   // 131072 threads
  int c = idx >> 9;          // 0..255  (column of R)
  int k = idx & 511;         // 0..511  (row of R)
  Rt[idx] = R[k * M_OUT + c];                        // Rt[c][k], row-major [256][512]
}

__global__ void k_gram(const float* __restrict__ R, float* __restrict__ G) {
  int i = blockIdx.x;        // 0..255
  int j = threadIdx.x;       // 0..255
  float s = 0.f;
  for (int k = 0; k < D_OUT; ++k)
    s = fmaf(R[k * M_OUT + i], R[k * M_OUT + j], s);
  G[i * M_OUT + j] = s;
}

// Power iteration on symmetric PSD G -> lambda_max = ||R||_2^2
__global__ void k_power(const float* __restrict__ G, float* __restrict__ eta) {
  __shared__ float v[256];
  __shared__ float red[256];
  int t = threadIdx.x;
  v[t] = 0.0625f;            // unit-norm start
  __syncthreads();
  float lam = 0.f;
  for (int it = 0; it < 96; ++it) {
    float w = 0.f;
    for (int j = 0; j < 256; ++j) w = fmaf(G[t * 256 + j], v[j], w);
    red[t] = w * w;
    __syncthreads();
    for (int s = 128; s > 0; s >>= 1) {
      if (t < s) red[t] += red[t + s];
      __syncthreads();
    }
    float n2 = red[0];
    lam = sqrtf(n2);
    float inv = 1.f / (lam + 1e-30f);
    __syncthreads();
    v[t] = w * inv;
    __syncthreads();
  }
  if (t == 0) {
    eta[0] = 1.f / (lam + 1e-6f);   // eta
    eta[1] = lam + 1e-6f;           // 1/eta
  }
}

// ------------------------------------------------- membership: all(x@P <= p)
__global__ void __launch_bounds__(128)
k_in_member(const float* __restrict__ X, const float* __restrict__ P,
            const float* __restrict__ p, int* __restrict__ inm) {
  const int lane = threadIdx.x & 31;
  const int wave = threadIdx.x >> 5;
  const int half = lane >> 4;
  const int l16  = lane & 15;
  const int rb = blockIdx.x * 64 + wave * 16;    // 16 rows per wave

  unsigned rowok = 0xFFFFu;
  const float* arow = X + (size_t)(rb + l16) * D_IN + 2 * half;
  for (int pass = 0; pass < 2; ++pass) {         // 2 x 256 columns
    v8f acc[16];
    const v8f zv = {0.f, 0.f, 0.f, 0.f, 0.f, 0.f, 0.f, 0.f};
#pragma unroll
    for (int nt = 0; nt < 16; ++nt) acc[nt] = zv;

    for (int k = 0; k < D_IN / 4; ++k) {
      v2f a = *(const v2f*)(arow + 4 * k);                 // A frag: row l16, K pair
      const float* bp = P + (4 * k + 2 * half) * M_IN + pass * 256 + l16;
#pragma unroll
      for (int nt = 0; nt < 16; ++nt) {
        v2f b;
        b.x = bp[nt * 16];
        b.y = bp[nt * 16 + M_IN];
        acc[nt] = wmma4(a, b, acc[nt]);
      }
    }
#pragma unroll
    for (int nt = 0; nt < 16; ++nt) {
      float pv = p[pass * 256 + nt * 16 + l16];
#pragma unroll
      for (int g = 0; g < 8; ++g) {
        unsigned m = __builtin_amdgcn_ballot_w32(acc[nt][g] <= pv);
        if ((m & 0xFFFFu) != 0xFFFFu) rowok &= ~(1u << g);        // rows 0..7
        if ((m >> 16)     != 0xFFFFu) rowok &= ~(1u << (g + 8));  // rows 8..15
      }
    }
  }
  if (lane < 16) inm[rb + lane] = (int)((rowok >> lane) & 1u);
}

// ------------------------- membership all(o@R <= r) AND emit c = o@R - r
__global__ void __launch_bounds__(128)
k_out_member(const float* __restrict__ O, const float* __restrict__ R,
             const float* __restrict__ r, int* __restrict__ outm,
             float* __restrict__ c) {
  const int lane = threadIdx.x & 31;
  const int wave = threadIdx.x >> 5;
  const int half = lane >> 4;
  const int l16  = lane & 15;
  const int rb = blockIdx.x * 64 + wave * 16;

  v8f acc[16];
  const v8f zv = {0.f, 0.f, 0.f, 0.f, 0.f, 0.f, 0.f, 0.f};
#pragma unroll
  for (int nt = 0; nt < 16; ++nt) acc[nt] = zv;

  const float* arow = O + (size_t)(rb + l16) * D_OUT + 2 * half;
  for (int k = 0; k < D_OUT / 4; ++k) {
    v2f a = *(const v2f*)(arow + 4 * k);
    const float* bp = R + (4 * k + 2 * half) * M_OUT + l16;
#pragma unroll
    for (int nt = 0; nt < 16; ++nt) {
      v2f b;
      b.x = bp[nt * 16];
      b.y = bp[nt * 16 + M_OUT];
      acc[nt] = wmma4(a, b, acc[nt]);
    }
  }

  unsigned rowok = 0xFFFFu;
#pragma unroll
  for (int nt = 0; nt < 16; ++nt) {
    float rv = r[nt * 16 + l16];
#pragma unroll
    for (int g = 0; g < 8; ++g) {
      float val = acc[nt][g];
      c[(size_t)(rb + g + 8 * half) * M_OUT + nt * 16 + l16] = val - rv;
      unsigned m = __builtin_amdgcn_ballot_w32(val <= rv);
      if ((m & 0xFFFFu) != 0xFFFFu) rowok &= ~(1u << g);
      if ((m >> 16)     != 0xFFFFu) rowok &= ~(1u << (g + 8));
    }
  }
  if (lane < 16) outm[rb + lane] = (int)((rowok >> lane) & 1u);
}

// ------------------------------------------------ projection iteration kernel
// LDS: G[256][256] (256KB) + per-wave lam[16][256] (4x16KB) = 320KB (WGP max).
__global__ void __launch_bounds__(128)
k_project(const float* __restrict__ O, const float* __restrict__ Gg,
          const float* __restrict__ Rt, const float* __restrict__ c,
          const int* __restrict__ inm, const int* __restrict__ outm,
          const float* __restrict__ etap, float* __restrict__ out) {
  extern __shared__ float smem[];
  float* Gs = smem;                                  // 65536 floats
  const int tid  = threadIdx.x;
  const int lane = tid & 31;
  const int wave = tid >> 5;
  const int half = lane >> 4;
  const int l16  = lane & 15;
  float* lam = smem + 65536 + wave * (16 * 256);     // row-major [16][256]
  const int rb = blockIdx.x * 64 + wave * 16;

  // Cooperative G load: async global->LDS (no VGPR round trip) when available.
#if HAVE_ASYNC_LDS
  {
    typedef __attribute__((address_space(1))) v4i_g* as1p;
    typedef __attribute__((address_space(3))) v4i_g* as3p;
    float* Gnc = const_cast<float*>(Gg);
    for (int i = tid * 4; i < 65536; i += 128 * 4) {
      v4i_g* g0 = reinterpret_cast<v4i_g*>(Gnc + i);
      v4i_g* l0 = reinterpret_cast<v4i_g*>(Gs + i);
      __builtin_amdgcn_global_load_async_to_lds_b128((as1p)g0, (as3p)l0, 0, 0);
    }
#  if __has_builtin(__builtin_amdgcn_s_wait_asynccnt)
    __builtin_amdgcn_s_wait_asynccnt(0);
#  else
    asm volatile("s_wait_asynccnt 0" ::: "memory");
#  endif
  }
#else
  for (int i = tid * 4; i < 65536; i += 128 * 4)
    *(v4f*)(Gs + i) = *(const v4f*)(Gg + i);
#endif
  __syncthreads();

  bool nd = false;
  if (lane < 16) {
    int row = rb + lane;
    nd = (inm[row] != 0) && (outm[row] == 0);
  }
  const unsigned needmask = __builtin_amdgcn_ballot_w32(nd) & 0xFFFFu;

  if (needmask) {
    const float eta     = etap[0];
    const float inv_eta = etap[1];

    for (int i = lane; i < 16 * 256; i += 32) lam[i] = 0.f;   // lam0 = 0 (LDS)
    asm volatile("s_wait_dscnt 0" ::: "memory");

    // c tile + lam live in registers, C-layout (lane l16 -> col, VGPR g -> row)
    v8f creg[16];
    v8f lamreg[16];
    const v8f zv = {0.f, 0.f, 0.f, 0.f, 0.f, 0.f, 0.f, 0.f};
#pragma unroll
    for (int nt = 0; nt < 16; ++nt) {
      lamreg[nt] = zv;
#pragma unroll
      for (int g = 0; g < 8; ++g)
        creg[nt][g] = c[(size_t)(rb + g + 8 * half) * M_OUT + nt * 16 + l16];
    }

    for (int it = 0; it < PROJ_ITERS; ++it) {
      // acc_init = -(lam/eta + c)  ->  lam_new = relu(-eta * (acc_init + lam@G))
      v8f acc[16];
#pragma unroll
      for (int nt = 0; nt < 16; ++nt)
#pragma unroll
        for (int g = 0; g < 8; ++g)
          acc[nt][g] = -(lamreg[nt][g] * inv_eta + creg[nt][g]);

      // t = lam @ G : A frags from per-wave lam LDS, B frags from shared G
      for (int k = 0; k < 64; ++k) {
        v2f a = *(const v2f*)(lam + l16 * 256 + 4 * k + 2 * half);
        const float* bp = Gs + (4 * k + 2 * half) * 256 + l16;
#pragma unroll
        for (int nt = 0; nt < 16; ++nt) {
          v2f b;
          b.x = bp[nt * 16];
          b.y = bp[nt * 16 + 256];
          acc[nt] = wmma4(a, b, acc[nt]);
        }
      }

      // lam = relu(-eta * acc): update registers + row-major LDS copy (A frags)
#pragma unroll
      for (int nt = 0; nt < 16; ++nt)
#pragma unroll
        for (int g = 0; g < 8; ++g) {
          float ln = fmaxf(-eta * acc[nt][g], 0.f);
          lamreg[nt][g] = ln;
          lam[(g + 8 * half) * 256 + nt * 16 + l16] = ln;
        }
      asm volatile("s_wait_dscnt 0" ::: "memory");
    }

    // Epilogue: y = o - lam @ R^T ; per-row select vs o
    v2f afr[64];
#pragma unroll
    for (int k = 0; k < 64; ++k)
      afr[k] = *(const v2f*)(lam + l16 * 256 + 4 * k + 2 * half);

    for (int nt2 = 0; nt2 < 32; ++nt2) {
      v8f acc2 = {0.f, 0.f, 0.f, 0.f, 0.f, 0.f, 0.f, 0.f};
#pragma unroll
      for (int k = 0; k < 64; ++k) {
        const float* bp = Rt + (4 * k + 2 * half) * D_OUT + nt2 * 16 + l16;
        v2f b;
        b.x = bp[0];
        b.y = bp[D_OUT];
        acc2 = wmma4(afr[k], b, acc2);
      }
#pragma unroll
      for (int g = 0; g < 8; ++g) {
        int row = g + 8 * half;
        int col = nt2 * 16 + l16;
        float ov = O[(size_t)(rb + row) * D_OUT + col];
        float res = ((needmask >> row) & 1u) ? (ov - acc2[g]) : ov;
        out[(size_t)(rb + row) * D_OUT + col] = res;
      }
    }
  } else {
    // No row in this tile needs projection: straight copy
    for (int i = lane; i < 16 * D_OUT; i += 32) {
      int row = i >> 9, col = i & 511;
      out[(size_t)(rb + row) * D_OUT + col] = O[(size_t)(rb + row) * D_OUT + col];
    }
  }
}

// -------------------------------------------------------------------- launch
extern "C" void kernel_launch(void* const* d_in, const int* in_sizes, int n_in,
                              void* d_out, int out_size, void* d_ws, size_t ws_size,
                              hipStream_t stream) {
  const float* X = (const float*)d_in[0];   // inputs       [16384,1024]
  const float* O = (const float*)d_in[1];   // outputs_true [16384,512]
  const float* P = (const float*)d_in[2];   // [1024,512]
  const float* p = (const float*)d_in[3];   // [512]
  const float* R = (const float*)d_in[4];   // [512,256]
  const float* r = (const float*)d_in[5];   // [256]
  float* out = (float*)d_out;

  // Workspace layout (bytes): G | Rt | c | inm | outm | eta  (~17.7 MB)
  char* ws = (char*)d_ws;
  float* G    = (float*)(ws + 0);          // 256 KB
  float* Rt   = (float*)(ws + 262144);     // 512 KB
  float* cbuf = (float*)(ws + 786432);     // 16 MB
  int*   inm  = (int*)(ws + 17563648);     // 64 KB
  int*   outm = (int*)(ws + 17629184);     // 64 KB
  float* eta  = (float*)(ws + 17694720);   // 8 B

  k_transpose_R<<<512, 256, 0, stream>>>(R, Rt);
  k_gram<<<256, 256, 0, stream>>>(R, G);
  k_power<<<1, 256, 0, stream>>>(G, eta);
  k_in_member<<<256, 128, 0, stream>>>(X, P, p, inm);
  k_out_member<<<256, 128, 0, stream>>>(O, R, r, outm, cbuf);
  k_project<<<256, 128, 327680, stream>>>(O, G, Rt, cbuf, inm, outm, eta, out);
}